// RoiPoolingConv_59579786330493
// MI455X (gfx1250) — compile-verified
//
#include <hip/hip_runtime.h>

// RoiPoolingConv (crop_and_resize bilinear), MI455X / gfx1250.
//
// Bandwidth-bound gather: 33.5MB feature map (L2-resident; L2=192MB) +
// 51.4MB streamed output -> ~3.7us HBM floor at 23.3 TB/s. No contraction
// dim -> WMMA inapplicable. Optimization targets: wave32-coalesced B128
// gathers, non-temporal stores (keep feature map hot in L2), and minimal
// VALU (no fp divides, no integer div/mod - all indices from launch geometry).

typedef float v4f __attribute__((ext_vector_type(4)));

#define POOL   7
#define FM_H   64
#define FM_W   64
#define FM_C   256
#define N_ROI  128
#define BATCH  8

// grid  = (7, 1024): blockIdx.x = iy, blockIdx.y = b*128 + roi
// block = 224 = 7 waves: wave id = ix (0..6), lane = channel chunk (8 ch each)
__global__ __launch_bounds__(224)
void roi_crop_resize_kernel(const float* __restrict__ fm,
                            const float* __restrict__ rois,
                            float* __restrict__ out) {
    const int lane = threadIdx.x & 31;   // channel chunk: c0 = lane*8
    const int ix   = threadIdx.x >> 5;   // wave index = output x (0..6)
    const int iy   = blockIdx.x;         // output y (0..6)
    const int br   = blockIdx.y;         // b*N_ROI + roi
    const int b    = br >> 7;

    // ROI = (x1_raw, y1_raw, x2_raw, y2_raw), block-uniform 16B load (L1-hot).
    const v4f rv = *reinterpret_cast<const v4f*>(rois + (br << 2));

    // Reference math: y1 = roi[1]/63 ; in_y = y1*63 + iy*((y2-y1)*63/6).
    // Divides replaced by reciprocal-constant multiplies (<=1 ulp, values
    // bounded by 63) to avoid the ~15-op IEEE v_div_scale sequences.
    const float inv63 = 0.015873015873015872f;  // 1/63
    const float inv6  = 0.16666666666666666f;   // 1/6
    const float fs    = 63.0f;
    const float y1 = rv.y * inv63;
    const float x1 = rv.x * inv63;
    const float y2 = rv.w * inv63;
    const float x2 = rv.z * inv63;

    const float in_y = y1 * fs + (float)iy * ((y2 - y1) * fs * inv6);
    const float in_x = x1 * fs + (float)ix * ((x2 - x1) * fs * inv6);

    const bool valid = (in_y >= 0.0f) && (in_y <= 63.0f) &&
                       (in_x >= 0.0f) && (in_x <= 63.0f);

    const float topf  = floorf(in_y);
    const float leftf = floorf(in_x);
    const float ylerp = in_y - topf;
    const float xlerp = in_x - leftf;

    const int ti = (int)fminf(fmaxf(topf,        0.0f), 63.0f);
    const int bi = (int)fminf(fmaxf(ceilf(in_y), 0.0f), 63.0f);
    const int li = (int)fminf(fmaxf(leftf,       0.0f), 63.0f);
    const int ri = (int)fminf(fmaxf(ceilf(in_x), 0.0f), 63.0f);

    const float* base = fm + (size_t)b * (FM_H * FM_W * FM_C);
    const int c0 = lane << 3;            // 8 channels per thread

    // Row offsets: ti*W*C = ti<<14, li*C = li<<8  (pure shifts, no mul).
    const int offT = (ti << 14) + c0;
    const int offB = (bi << 14) + c0;
    const int offL = (li << 8);
    const int offR = (ri << 8);

    // 4 corners x 2 adjacent float4 -> each wave streams the full 1KB row
    // contiguously (32 lanes x 32B). All repeats across positions hit L2.
    const v4f* ptl = reinterpret_cast<const v4f*>(base + offT + offL);
    const v4f* ptr = reinterpret_cast<const v4f*>(base + offT + offR);
    const v4f* pbl = reinterpret_cast<const v4f*>(base + offB + offL);
    const v4f* pbr = reinterpret_cast<const v4f*>(base + offB + offR);

    const v4f tl0 = ptl[0], tl1 = ptl[1];
    const v4f tr0 = ptr[0], tr1 = ptr[1];
    const v4f bl0 = pbl[0], bl1 = pbl[1];
    const v4f br0 = pbr[0], br1 = pbr[1];

    const v4f top0 = tl0 + (tr0 - tl0) * xlerp;
    const v4f top1 = tl1 + (tr1 - tl1) * xlerp;
    const v4f bot0 = bl0 + (br0 - bl0) * xlerp;
    const v4f bot1 = bl1 + (br1 - bl1) * xlerp;
    v4f res0 = top0 + (bot0 - top0) * ylerp;
    v4f res1 = top1 + (bot1 - top1) * ylerp;
    if (!valid) {
        res0 = (v4f){0.0f, 0.0f, 0.0f, 0.0f};
        res1 = (v4f){0.0f, 0.0f, 0.0f, 0.0f};
    }

    // Output written once, never re-read: non-temporal (th:NT) stores so the
    // 51MB stream doesn't evict the feature map from L2.
    const size_t opos = (size_t)br * (POOL * POOL) + iy * POOL + ix;
    v4f* dst = reinterpret_cast<v4f*>(out + (opos << 8) + c0);
    __builtin_nontemporal_store(res0, dst);
    __builtin_nontemporal_store(res1, dst + 1);
}

extern "C" void kernel_launch(void* const* d_in, const int* in_sizes, int n_in,
                              void* d_out, int out_size, void* d_ws, size_t ws_size,
                              hipStream_t stream) {
    (void)in_sizes; (void)n_in; (void)out_size; (void)d_ws; (void)ws_size;
    const float* fm   = (const float*)d_in[0];   // [8,64,64,256]   f32
    const float* rois = (const float*)d_in[1];   // [8,128,4]       f32
    float* out        = (float*)d_out;           // [8,128,7,7,256] f32

    dim3 grid(POOL, BATCH * N_ROI);              // (7, 1024)
    dim3 block(POOL * 32);                       // 224 threads = 7 waves
    roi_crop_resize_kernel<<<grid, block, 0, stream>>>(fm, rois, out);
}